// ScalarBorn_40114994544663
// MI455X (gfx1250) — compile-verified
//
#include <hip/hip_runtime.h>
#include <cstdint>

// ---------------- problem constants (from reference) ----------------
#define NSHOTS 4
#define NY 344
#define NX 344
#define NT 40
#define NSRC 1
#define NREC 300
#define FD_PAD 2

#define DT 1e-3f
#define INV_H 0.25f        // 1/4.0
#define INV_H2 0.0625f     // 1/16.0

#define S2 (NY * NX)                         // one shot-plane
#define SFIELD ((size_t)NSHOTS * (size_t)S2) // one field (all shots)

// tile geometry for the step kernel
#define TX 32
#define TY 8
#define HX (TX + 4)
#define HY (TY + 4)
#define NBY (NY / TY)      // 43 stencil block-rows; block-row NBY does recording

// ---------------- CDNA5 async global->LDS helpers ----------------
__device__ __forceinline__ void async_ld_f32(const float* g, float* l) {
#if __has_builtin(__builtin_amdgcn_global_load_async_to_lds_b32)
  __builtin_amdgcn_global_load_async_to_lds_b32(
      (__attribute__((address_space(1))) int*)(uintptr_t)g,
      (__attribute__((address_space(3))) int*)(uintptr_t)l,
      /*offset=*/0, /*cpol=*/0);
#else
  // VGLOBAL async load: VDST = LDS byte address (low 32 bits of generic ptr),
  // VADDR = 64-bit global address, GV mode.
  asm volatile("global_load_async_to_lds_b32 %0, %1, off"
               :: "v"((unsigned)(uintptr_t)l), "v"(g)
               : "memory");
#endif
}

__device__ __forceinline__ void wait_async0() {
#if __has_builtin(__builtin_amdgcn_s_wait_asynccnt)
  __builtin_amdgcn_s_wait_asynccnt(0);
#else
  asm volatile("s_wait_asynccnt 0" ::: "memory");
#endif
}

// ---------------- one Born time step ----------------
// Fused: receiver sampling (extra block-row, reads cur only) + FD/PML step
// (writes nxt only) + point-source injection. No cross-block ordering hazard
// within one launch since readers and writers touch disjoint buffers.
__global__ __launch_bounds__(TX * TY) void k_step(
    const float* __restrict__ cur, float* __restrict__ nxt,
    const float* __restrict__ v, const float* __restrict__ scat,
    const float* __restrict__ ay, const float* __restrict__ by,
    const float* __restrict__ dby,
    const float* __restrict__ ax, const float* __restrict__ bx,
    const float* __restrict__ dbx,
    const int* __restrict__ src_i,
    const float* __restrict__ samp, const float* __restrict__ sampsc,
    const int* __restrict__ rec_i, const int* __restrict__ recsc_i,
    float* __restrict__ rec_out, float* __restrict__ recsc_out,
    int t) {
  __shared__ float su[HY][HX];   // wfc tile + halo
  __shared__ float ss[HY][HX];   // wfcsc tile + halo

  const int sh = blockIdx.z;
  const int tid = threadIdx.y * TX + threadIdx.x;

  const float* wfc    = cur + 0 * SFIELD + (size_t)sh * S2;
  const float* wfcsc  = cur + 6 * SFIELD + (size_t)sh * S2;

  // ---- fused receiver sampling (records wf state BEFORE this step) ----
  if (blockIdx.y == NBY) {
    int i = blockIdx.x * (TX * TY) + tid;   // receiver index within shot
    if (i < NREC) {
      rec_out[sh * NREC + i]   = wfc[rec_i[sh * NREC + i]];
      recsc_out[sh * NREC + i] = wfcsc[recsc_i[sh * NREC + i]];
    }
    return;   // whole block takes this branch; no barrier divergence
  }

  const float* wfp    = cur + 1 * SFIELD + (size_t)sh * S2;
  const float* psi0   = cur + 2 * SFIELD + (size_t)sh * S2;
  const float* psi1   = cur + 3 * SFIELD + (size_t)sh * S2;
  const float* zeta0  = cur + 4 * SFIELD + (size_t)sh * S2;
  const float* zeta1  = cur + 5 * SFIELD + (size_t)sh * S2;
  const float* wfpsc  = cur + 7 * SFIELD + (size_t)sh * S2;
  const float* psisc0 = cur + 8 * SFIELD + (size_t)sh * S2;
  const float* psisc1 = cur + 9 * SFIELD + (size_t)sh * S2;
  const float* zetasc0 = cur + 10 * SFIELD + (size_t)sh * S2;
  const float* zetasc1 = cur + 11 * SFIELD + (size_t)sh * S2;

  const int gx0 = blockIdx.x * TX - 2;
  const int gy0 = blockIdx.y * TY - 2;

  // Stage halo'd wfc / wfcsc tiles into LDS via async loads; out-of-domain
  // cells are zero-filled (matches jnp.pad zero boundary).
  for (int i = tid; i < HX * HY; i += TX * TY) {
    int lyy = i / HX, lxx = i % HX;
    int gy = gy0 + lyy, gx = gx0 + lxx;
    float* lu = &su[0][0] + i;
    float* ls = &ss[0][0] + i;
    if ((unsigned)gy < (unsigned)NY && (unsigned)gx < (unsigned)NX) {
      async_ld_f32(wfc + gy * NX + gx, lu);
      async_ld_f32(wfcsc + gy * NX + gx, ls);
    } else {
      *lu = 0.0f;
      *ls = 0.0f;
    }
  }
  wait_async0();
  __syncthreads();

  const int x = blockIdx.x * TX + threadIdx.x;
  const int y = blockIdx.y * TY + threadIdx.y;
  if (x >= NX) return;  // y always in range (43*8 == 344); no barriers below

  const int lx = threadIdx.x + 2;
  const int ly = threadIdx.y + 2;

  const float C1 = 2.0f / 3.0f;    // diff1 coefficients
  const float C2 = 1.0f / 12.0f;
  const float D2A = -1.0f / 12.0f; // diff2 coefficients
  const float D2B = 4.0f / 3.0f;

  // ---- background field derivatives from LDS ----
  float uc   = su[ly][lx];
  float um1y = su[ly - 1][lx], up1y = su[ly + 1][lx];
  float um2y = su[ly - 2][lx], up2y = su[ly + 2][lx];
  float um1x = su[ly][lx - 1], up1x = su[ly][lx + 1];
  float um2x = su[ly][lx - 2], up2x = su[ly][lx + 2];
  float dudy   = (C2 * um2y - C1 * um1y + C1 * up1y - C2 * up2y) * INV_H;
  float dudx   = (C2 * um2x - C1 * um1x + C1 * up1x - C2 * up2x) * INV_H;
  float d2udy2 = ((um2y + up2y) * D2A + (um1y + up1y) * D2B - 2.5f * uc) * INV_H2;
  float d2udx2 = ((um2x + up2x) * D2A + (um1x + up1x) * D2B - 2.5f * uc) * INV_H2;

  // ---- scattered field derivatives from LDS ----
  float sc   = ss[ly][lx];
  float sm1y = ss[ly - 1][lx], sp1y = ss[ly + 1][lx];
  float sm2y = ss[ly - 2][lx], sp2y = ss[ly + 2][lx];
  float sm1x = ss[ly][lx - 1], sp1x = ss[ly][lx + 1];
  float sm2x = ss[ly][lx - 2], sp2x = ss[ly][lx + 2];
  float dscdy   = (C2 * sm2y - C1 * sm1y + C1 * sp1y - C2 * sp2y) * INV_H;
  float dscdx   = (C2 * sm2x - C1 * sm1x + C1 * sp1x - C2 * sp2x) * INV_H;
  float d2scdy2 = ((sm2y + sp2y) * D2A + (sm1y + sp1y) * D2B - 2.5f * sc) * INV_H2;
  float d2scdx2 = ((sm2x + sp2x) * D2A + (sm1x + sp1x) * D2B - 2.5f * sc) * INV_H2;

  const int idx = y * NX + x;
  const size_t gi = (size_t)idx;

  float ayv = ay[y], byv = by[y], dbyv = dby[y];
  float axv = ax[x], bxv = bx[x], dbxv = dbx[x];

  // ---- diff1 of (a * psi) along each axis (zero beyond boundary) ----
  float apy = 0.0f, apysc = 0.0f;
  if (y >= 2)     { apy += C2 * ay[y - 2] * psi0[(y - 2) * NX + x];
                    apysc += C2 * ay[y - 2] * psisc0[(y - 2) * NX + x]; }
  if (y >= 1)     { apy -= C1 * ay[y - 1] * psi0[(y - 1) * NX + x];
                    apysc -= C1 * ay[y - 1] * psisc0[(y - 1) * NX + x]; }
  if (y + 1 < NY) { apy += C1 * ay[y + 1] * psi0[(y + 1) * NX + x];
                    apysc += C1 * ay[y + 1] * psisc0[(y + 1) * NX + x]; }
  if (y + 2 < NY) { apy -= C2 * ay[y + 2] * psi0[(y + 2) * NX + x];
                    apysc -= C2 * ay[y + 2] * psisc0[(y + 2) * NX + x]; }
  apy *= INV_H; apysc *= INV_H;

  float apx = 0.0f, apxsc = 0.0f;
  if (x >= 2)     { apx += C2 * ax[x - 2] * psi1[y * NX + x - 2];
                    apxsc += C2 * ax[x - 2] * psisc1[y * NX + x - 2]; }
  if (x >= 1)     { apx -= C1 * ax[x - 1] * psi1[y * NX + x - 1];
                    apxsc -= C1 * ax[x - 1] * psisc1[y * NX + x - 1]; }
  if (x + 1 < NX) { apx += C1 * ax[x + 1] * psi1[y * NX + x + 1];
                    apxsc += C1 * ax[x + 1] * psisc1[y * NX + x + 1]; }
  if (x + 2 < NX) { apx -= C2 * ax[x + 2] * psi1[y * NX + x + 2];
                    apxsc -= C2 * ax[x + 2] * psisc1[y * NX + x + 2]; }
  apx *= INV_H; apxsc *= INV_H;

  float p0 = psi0[gi], p1 = psi1[gi], z0 = zeta0[gi], z1 = zeta1[gi];
  float p0s = psisc0[gi], p1s = psisc1[gi], z0s = zetasc0[gi], z1s = zetasc1[gi];

  float tmp0  = (1.0f + byv) * d2udy2 + dbyv * dudy + apy;
  float tmp1  = (1.0f + bxv) * d2udx2 + dbxv * dudx + apx;
  float tmp0s = (1.0f + byv) * d2scdy2 + dbyv * dscdy + apysc;
  float tmp1s = (1.0f + bxv) * d2scdx2 + dbxv * dscdx + apxsc;

  float w_sum  = (1.0f + byv) * tmp0 + ayv * z0 + (1.0f + bxv) * tmp1 + axv * z1;
  float ws_sum = (1.0f + byv) * tmp0s + ayv * z0s + (1.0f + bxv) * tmp1s + axv * z1s;

  float vv = v[idx];            // v, scatter have shape (1, NY, NX): shared over shots
  float v2dt2 = vv * vv * (DT * DT);
  float n_wfc = v2dt2 * w_sum + 2.0f * uc - wfp[gi];
  float n_wfcsc = v2dt2 * ws_sum + 2.0f * sc - wfpsc[gi]
                + 2.0f * vv * scat[idx] * (DT * DT) * w_sum;

  // fused point-source injection (NSRC == 1)
  if (idx == src_i[sh]) {
    n_wfc   += samp[t * NSHOTS + sh];
    n_wfcsc += sampsc[t * NSHOTS + sh];
  }

  float* out = nxt + (size_t)sh * S2;
  out[0 * SFIELD + gi]  = n_wfc;
  out[1 * SFIELD + gi]  = uc;
  out[2 * SFIELD + gi]  = byv * dudy + ayv * p0;
  out[3 * SFIELD + gi]  = bxv * dudx + axv * p1;
  out[4 * SFIELD + gi]  = byv * tmp0 + ayv * z0;
  out[5 * SFIELD + gi]  = bxv * tmp1 + axv * z1;
  out[6 * SFIELD + gi]  = n_wfcsc;
  out[7 * SFIELD + gi]  = sc;
  out[8 * SFIELD + gi]  = byv * dscdy + ayv * p0s;
  out[9 * SFIELD + gi]  = bxv * dscdx + axv * p1s;
  out[10 * SFIELD + gi] = byv * tmp0s + ayv * z0s;
  out[11 * SFIELD + gi] = bxv * tmp1s + axv * z1s;
}

// ---------------- final interior copy of wfc / wfcsc ----------------
__global__ void k_final(const float* __restrict__ cur,
                        float* __restrict__ o_wfc,
                        float* __restrict__ o_wfcsc) {
  const int NIY = NY - 2 * FD_PAD;
  const int NIX = NX - 2 * FD_PAD;
  int i = blockIdx.x * blockDim.x + threadIdx.x;
  if (i >= NSHOTS * NIY * NIX) return;
  int sh = i / (NIY * NIX);
  int r = i % (NIY * NIX);
  int y = r / NIX + FD_PAD;
  int x = r % NIX + FD_PAD;
  size_t gi = (size_t)sh * S2 + (size_t)y * NX + x;
  o_wfc[i]   = cur[0 * SFIELD + gi];
  o_wfcsc[i] = cur[6 * SFIELD + gi];
}

// ---------------- host entry ----------------
extern "C" void kernel_launch(void* const* d_in, const int* in_sizes, int n_in,
                              void* d_out, int out_size, void* d_ws, size_t ws_size,
                              hipStream_t stream) {
  const float* v       = (const float*)d_in[0];
  const float* scat    = (const float*)d_in[1];
  const float* samp    = (const float*)d_in[2];
  const float* sampsc  = (const float*)d_in[3];
  const float* a_y     = (const float*)d_in[4];
  const float* b_y     = (const float*)d_in[5];
  const float* dbdy    = (const float*)d_in[6];
  const float* a_x     = (const float*)d_in[7];
  const float* b_x     = (const float*)d_in[8];
  const float* dbdx    = (const float*)d_in[9];
  const int*   src_i   = (const int*)d_in[10];
  const int*   rec_i   = (const int*)d_in[11];
  const int*   recsc_i = (const int*)d_in[12];

  float* ws = (float*)d_ws;
  float* bufA = ws;                 // 12 fields
  float* bufB = ws + 12 * SFIELD;   // 12 fields
  // zero both ping-pong buffers (wf0 starts at zero; deterministic per call)
  (void)hipMemsetAsync(d_ws, 0, 24 * SFIELD * sizeof(float), stream);

  float* rec      = (float*)d_out;                       // (NT, NSHOTS, NREC)
  float* recsc    = rec + (size_t)NT * NSHOTS * NREC;    // (NT, NSHOTS, NREC)
  float* o_wfc    = recsc + (size_t)NT * NSHOTS * NREC;  // (NSHOTS, 340, 340)
  float* o_wfcsc  = o_wfc + (size_t)NSHOTS * (NY - 4) * (NX - 4);

  // grid.y has one extra block-row (blockIdx.y == NBY) for receiver sampling
  dim3 sgrid((NX + TX - 1) / TX, NBY + 1, NSHOTS);
  dim3 sblock(TX, TY);
  int fin_threads = NSHOTS * (NY - 4) * (NX - 4);

  float* cur = bufA;
  float* nxt = bufB;
  for (int t = 0; t < NT; ++t) {
    k_step<<<sgrid, sblock, 0, stream>>>(
        cur, nxt, v, scat, a_y, b_y, dbdy, a_x, b_x, dbdx,
        src_i, samp, sampsc,
        rec_i, recsc_i,
        rec + (size_t)t * NSHOTS * NREC,
        recsc + (size_t)t * NSHOTS * NREC,
        t);
    float* tswap = cur; cur = nxt; nxt = tswap;
  }
  k_final<<<(fin_threads + 255) / 256, 256, 0, stream>>>(cur, o_wfc, o_wfcsc);
}